// Attention_48137993454135
// MI455X (gfx1250) — compile-verified
//
#include <hip/hip_runtime.h>

// CDNA5 / gfx1250 wave32 attention using V_WMMA_F32_16X16X4_F32 (exact fp32 path).
// Phase 1: flash attention, one wave per (n, head64, q-tile16) -> d_ws (8 MB).
// Phase 2: projection GEMM y = X @ W^T + b, one wave per 16x16 output tile.

typedef float v2f __attribute__((ext_vector_type(2)));
typedef float v4f __attribute__((ext_vector_type(4)));
typedef float v8f __attribute__((ext_vector_type(8)));
typedef int   v4i __attribute__((ext_vector_type(4)));

#define SEQ  1024
#define EMB  1024
#define NH   64     // effective heads (reference's reshape quirk)
#define DH   16     // effective head dim

// ---------------------------------------------------------------------------
// Phase 1: per-wave flash attention on 16 query rows of one (n, h).
// Computes S^T = K x Q^T so the softmax (key) axis lies on the M/VGPR dim:
// row stats reduce with 8 in-lane ops + one shfl_xor(16), and become
// lane-uniform scalars (q = lane&15). O^T = V^T x P^T accumulates in-register.
// ---------------------------------------------------------------------------
__global__ __launch_bounds__(32) void attn_flash_wmma(
    const float* __restrict__ keys, const float* __restrict__ query,
    const float* __restrict__ values, const int* __restrict__ mask,
    float* __restrict__ attn_out)
{
    const int lane = threadIdx.x;        // 0..31, one full wave per block
    const int q    = lane & 15;          // column index (query within tile)
    const int half = lane >> 4;          // 0: lanes 0-15, 1: lanes 16-31

    const int idx = blockIdx.x;
    const int qt  = idx & 63;            // q-tile (64 tiles of 16 rows)
    const int h   = (idx >> 6) & 63;     // head 0..63
    const int n   = idx >> 12;           // batch 0..1
    const int q0  = qt * 16;

    // B operand for scores: B[d][q] = Q[q0+q][h*16+d]; per chunk dd, lane
    // holds d = 4*dd + 2*half {+0,+1}  -> contiguous float2, loaded once.
    const float* Qrow = query + ((size_t)(n * SEQ + q0 + q)) * EMB + h * DH;
    v2f Bq[4];
#pragma unroll
    for (int dd = 0; dd < 4; ++dd)
        Bq[dd] = *reinterpret_cast<const v2f*>(Qrow + dd * 4 + 2 * half);

    v8f   o    = {};        // O^T accum: o[r] = O^T[d = r + 8*half][q]
    float mold = -1e30f;    // running row max (per-lane, lane-uniform in q)
    float lsum = 0.0f;      // running row sum

    const float* Kbase = keys   + ((size_t)(n * SEQ)) * EMB + h * DH;
    const float* Vbase = values + ((size_t)(n * SEQ)) * EMB + h * DH;
    const int*   Mrow  = mask + (size_t)n * SEQ * SEQ + (size_t)(q0 + q) * SEQ;

    for (int kt = 0; kt < SEQ / 16; ++kt) {
        // ---- S^T tile = K(16k x 16d) x Q^T(16d x 16q), 4 chained K=4 WMMAs
        const float* Krow = Kbase + (size_t)(kt * 16 + q) * EMB;  // M = key = lane&15
        v8f s = {};
#pragma unroll
        for (int dd = 0; dd < 4; ++dd) {
            v2f a = *reinterpret_cast<const v2f*>(Krow + dd * 4 + 2 * half);
            s = __builtin_amdgcn_wmma_f32_16x16x4_f32(
                    false, a, false, Bq[dd], (short)0, s, false, false);
        }

        // ---- mask (element (kRow=r+8*half, q)) + 1/sqrt(E) scale
        const int* mb = Mrow + kt * 16 + 8 * half;   // 8 contiguous ints
        v4i m0 = *reinterpret_cast<const v4i*>(mb);
        v4i m1 = *reinterpret_cast<const v4i*>(mb + 4);
        float p[8];
#pragma unroll
        for (int r = 0; r < 8; ++r) {
            float sr = s[r];
            int   mv = (r < 4) ? m0[r] : m1[r - 4];
            sr   = (mv == 0) ? -1e20f : sr;
            p[r] = sr * 0.03125f;                    // 1/32
        }

        // ---- flash softmax update (per-lane scalars; q fixed per lane)
        float tmax = p[0];
#pragma unroll
        for (int r = 1; r < 8; ++r) tmax = fmaxf(tmax, p[r]);
        tmax = fmaxf(tmax, __shfl_xor(tmax, 16));
        const float mnew  = fmaxf(mold, tmax);
        const float alpha = __expf(mold - mnew);
        float tsum = 0.0f;
#pragma unroll
        for (int r = 0; r < 8; ++r) { p[r] = __expf(p[r] - mnew); tsum += p[r]; }
        tsum += __shfl_xor(tsum, 16);
        lsum = lsum * alpha + tsum;
        mold = mnew;
#pragma unroll
        for (int r = 0; r < 8; ++r) o[r] *= alpha;   // lane-uniform rescale

        // ---- O^T += V^T(16d x k) x P^T(k x 16q)
        const float* Vt = Vbase + (size_t)(kt * 16) * EMB + q;   // d = lane&15
#pragma unroll
        for (int kk = 0; kk < 4; ++kk) {
            const int k0 = 4 * kk + 2 * half;        // this lane's A/B K index
            v2f av;                                   // A = V^T: (d, k0), (d, k0+1)
            av.x = Vt[(size_t)k0 * EMB];
            av.y = Vt[(size_t)(k0 + 1) * EMB];
            // restripe P^T from C-layout into B-layout:
            // value P^T[k0][q] lives at lane (q + 16*(k0>>3)), register k0&7
            const int a       = (kk & 1) * 4;        // = (4*kk) & 7
            const int srcLane = q | ((kk >> 1) << 4);
            float t0 = __shfl(p[a],     srcLane);
            float t2 = __shfl(p[a + 2], srcLane);
            float t1 = __shfl(p[a + 1], srcLane);
            float t3 = __shfl(p[a + 3], srcLane);
            v2f bp;
            bp.x = half ? t2 : t0;
            bp.y = half ? t3 : t1;
            o = __builtin_amdgcn_wmma_f32_16x16x4_f32(
                    false, av, false, bp, (short)0, o, false, false);
        }
    }

    // ---- normalize and store O (lane writes 8 contiguous d values)
    const float rl = 1.0f / lsum;
    float* Orow = attn_out + ((size_t)(n * SEQ + q0 + q)) * EMB + h * DH + 8 * half;
    v4f o0, o1;
#pragma unroll
    for (int r = 0; r < 4; ++r) { o0[r] = o[r] * rl; o1[r] = o[r + 4] * rl; }
    *reinterpret_cast<v4f*>(Orow)     = o0;
    *reinterpret_cast<v4f*>(Orow + 4) = o1;
}

// ---------------------------------------------------------------------------
// Phase 2: y = X @ W^T + b.  One wave per 16x16 output tile; 256 chained
// K=4 f32 WMMAs over the 1024-deep contraction. A rows (X) and B rows (W,
// since y uses W^T) are both contiguous float2 loads in the A/B lane maps.
// ---------------------------------------------------------------------------
__global__ __launch_bounds__(32) void proj_wmma(
    const float* __restrict__ X, const float* __restrict__ W,
    const float* __restrict__ bvec, float* __restrict__ out)
{
    const int lane = threadIdx.x;
    const int q    = lane & 15;
    const int half = lane >> 4;
    const int j0 = (blockIdx.x & 63) * 16;   // output column tile
    const int i0 = (blockIdx.x >> 6) * 16;   // output row tile (rows of N*S)

    const float* Xrow = X + (size_t)(i0 + q) * EMB;  // A: M = i = lane&15
    const float* Wrow = W + (size_t)(j0 + q) * EMB;  // B: N = j = lane&15
    v8f acc = {};
#pragma unroll 4
    for (int ee = 0; ee < EMB / 4; ++ee) {
        v2f a = *reinterpret_cast<const v2f*>(Xrow + ee * 4 + 2 * half);
        v2f b = *reinterpret_cast<const v2f*>(Wrow + ee * 4 + 2 * half);
        acc = __builtin_amdgcn_wmma_f32_16x16x4_f32(
                  false, a, false, b, (short)0, acc, false, false);
    }
    const float bj = bvec[j0 + q];           // per-lane bias (column j)
#pragma unroll
    for (int r = 0; r < 8; ++r)
        out[(size_t)(i0 + r + 8 * half) * EMB + j0 + q] = acc[r] + bj;
}

// ---------------------------------------------------------------------------
extern "C" void kernel_launch(void* const* d_in, const int* in_sizes, int n_in,
                              void* d_out, int out_size, void* d_ws, size_t ws_size,
                              hipStream_t stream)
{
    const float* keys   = (const float*)d_in[0];
    const float* query  = (const float*)d_in[1];
    const float* values = (const float*)d_in[2];
    const int*   mask   = (const int*)d_in[3];
    const float* W_out  = (const float*)d_in[4];
    const float* b_out  = (const float*)d_in[5];
    float* out  = (float*)d_out;
    float* attn = (float*)d_ws;              // N*S*E floats = 8 MB scratch

    const int N = 2;
    // Phase 1: N * 64 heads * 64 q-tiles waves
    attn_flash_wmma<<<N * NH * (SEQ / 16), 32, 0, stream>>>(
        keys, query, values, mask, attn);
    // Phase 2: (N*S/16) * (E/16) output tiles
    proj_wmma<<<(N * SEQ / 16) * (EMB / 16), 32, 0, stream>>>(
        attn, W_out, b_out, out);
}